// TiThTe_1958505087094
// MI455X (gfx1250) — compile-verified
//
#include <hip/hip_runtime.h>

// ---------------------------------------------------------------------------
// 3-state RC thermal network scan (B independent buildings, T sequential steps)
//   state: [T_in, T_heater, T_env] per building
//   per-step forcing (shared by all buildings): [T_out, heatPower, solarGains]
// Output: full trajectory (T, B, 3) fp32  ->  ~430 MB streaming stores.
//
// MI455X strategy:
//   * one thread per building, state+coeffs in registers (latency-chain bound)
//   * forcing data staged to LDS with double-buffered ASYNC global->LDS DMA
//     (gfx1250 ASYNCcnt path), s_wait_asynccnt pipelining
//   * x4 step unroll -> LDS reads fuse to ds_load_b128, less loop overhead
//   * non-temporal stores for the output stream (never re-read; 23.3 TB/s path)
// ---------------------------------------------------------------------------

#define NTHREADS 256
#define CHUNK    584                 // timesteps per LDS chunk (8760 = 15*584)
#define CHUNK_F  (CHUNK * 3)         // 1752 floats per chunk
#define CHUNK_Q  (CHUNK_F / 4)       // 438 float4 quads per chunk
#define ISSUES   ((CHUNK_Q + NTHREADS - 1) / NTHREADS)   // = 2 b128 issues/wave

#define TAME_V  100000.0f
#define TAME_SQ 1.0e10f              // TAME*TAME (exact fp32 rounding matches ref)

// 16-byte vector types with explicit address spaces (builtin signature wants
// 'int __vector(4) addrspace(1)*' per clang's diagnostic).
typedef int v4i __attribute__((vector_size(16)));
typedef __attribute__((address_space(1))) v4i gv4i;   // global
typedef __attribute__((address_space(3))) v4i lv4i;   // LDS

// ---- CDNA5 async global -> LDS copy (tracked with ASYNCcnt) ---------------
__device__ __forceinline__ void async_g2l_b128(const float* g, const float* l) {
#if __has_builtin(__builtin_amdgcn_global_load_async_to_lds_b128)
  __builtin_amdgcn_global_load_async_to_lds_b128(
      (gv4i*)g, (lv4i*)l, /*offset=*/0, /*cpol=*/0);
#else
  unsigned loff = (unsigned)(unsigned long long)
      (__attribute__((address_space(3))) const void*)l;
  asm volatile("global_load_async_to_lds_b128 %0, %1, off"
               :: "v"(loff), "v"(g) : "memory");
#endif
}

__device__ __forceinline__ void wait_asynccnt_le_issues() {
#if __has_builtin(__builtin_amdgcn_s_wait_asynccnt)
  __builtin_amdgcn_s_wait_asynccnt(ISSUES);
#else
  asm volatile("s_wait_asynccnt 2" ::: "memory");
#endif
}

__device__ __forceinline__ void wait_asynccnt_zero() {
#if __has_builtin(__builtin_amdgcn_s_wait_asynccnt)
  __builtin_amdgcn_s_wait_asynccnt(0);
#else
  asm volatile("s_wait_asynccnt 0" ::: "memory");
#endif
}

// tame(x) = x * 1e5 / sqrt(x*x + 1e10)  -> fma + v_rsq_f32 + mul (mul overlaps)
__device__ __forceinline__ float tame(float x) {
  float r = __builtin_amdgcn_rsqf(__builtin_fmaf(x, x, TAME_SQ));
  return (x * TAME_V) * r;
}

__global__ __launch_bounds__(NTHREADS, 4)
void rc3_scan_kernel(const float* __restrict__ state0,   // (B,3)
                     const float* __restrict__ params,   // (B,6) C1,R1,C2,R2,C3,R3
                     const float* __restrict__ ext,      // (T,3) Tout,heat,solar
                     const int*   __restrict__ dtp,      // scalar dt (int)
                     float* __restrict__ out,            // (T,B,3)
                     int B, int T)
{
  __shared__ __align__(16) float s_ext[2][CHUNK_F];

  const int  tid    = threadIdx.x;
  const int  b      = blockIdx.x * NTHREADS + tid;
  const bool active = (b < B);
  const float dtf   = (float)(*dtp);

  // Per-building state + coefficients in registers. Inactive lanes get zero
  // coefficients so the (unpredicated) step math stays finite.
  float Tin = 290.0f, Th = 290.0f, Te = 290.0f;
  float iR1 = 0.0f, iR2 = 0.0f, iR3 = 0.0f;
  float a1 = 0.0f, a2 = 0.0f, a3 = 0.0f;
  if (active) {
    Tin = state0[b * 3 + 0];
    Th  = state0[b * 3 + 1];
    Te  = state0[b * 3 + 2];
    const float C1 = params[b * 6 + 0], R1 = params[b * 6 + 1];
    const float C2 = params[b * 6 + 2], R2 = params[b * 6 + 3];
    const float C3 = params[b * 6 + 4], R3 = params[b * 6 + 5];
    iR1 = 1.0f / R1;  iR2 = 1.0f / R2;  iR3 = 1.0f / R3;
    a1  = dtf / C1;   a2  = dtf / C2;   a3  = dtf / C3;
  }

  const int totQ   = (T * 3) >> 2;                 // total float4 quads in ext
  const int nchunk = (T + CHUNK - 1) / CHUNK;

  // Kick off DMA of chunk 0 into buffer 0 (uniform ISSUES per wave, full EXEC;
  // indices are clamped so every wave issues exactly ISSUES instructions).
  {
#pragma unroll
    for (int i = 0; i < ISSUES; ++i) {
      int q  = tid + i * NTHREADS; if (q  > CHUNK_Q - 1) q  = CHUNK_Q - 1;
      int gq = q;                  if (gq > totQ    - 1) gq = totQ    - 1;
      async_g2l_b128(ext + 4 * gq, &s_ext[0][4 * q]);
    }
  }

  float* outp = out + (size_t)b * 3;               // advances by 3*B per step
  const size_t ostride = (size_t)B * 3;

  for (int c = 0; c < nchunk; ++c) {
    const int buf = c & 1;

    if (c + 1 < nchunk) {
      // Prefetch next chunk into the other buffer, then drain down to the
      // ISSUES just-issued ops => current chunk's DMA is complete.
      const int gbase = (c + 1) * CHUNK_Q;
#pragma unroll
      for (int i = 0; i < ISSUES; ++i) {
        int q  = tid + i * NTHREADS; if (q  > CHUNK_Q - 1) q  = CHUNK_Q - 1;
        int gq = gbase + q;          if (gq > totQ    - 1) gq = totQ    - 1;
        async_g2l_b128(ext + 4 * gq, &s_ext[buf ^ 1][4 * q]);
      }
      wait_asynccnt_le_issues();
    } else {
      wait_asynccnt_zero();
    }
    __syncthreads();   // all waves' DMA for this buffer visible to everyone

    const int steps = (T - c * CHUNK < CHUNK) ? (T - c * CHUNK) : CHUNK;
    const float* sp = s_ext[buf];

    // x4 unroll: 4 steps read 48 contiguous LDS bytes, 16B-aligned at the
    // unroll boundary -> fuses into 3x ds_load_b128 per 4 steps.
#pragma unroll 4
    for (int i = 0; i < steps; ++i) {
      const float Toutv = sp[3 * i + 0];   // LDS broadcast (same addr all lanes)
      const float heat  = sp[3 * i + 1];
      const float sol   = sp[3 * i + 2];

      const float Pw   = (Tin - Te) * iR1;          // P_in2Wall
      const float Ph   = (Tin - Th) * iR2;          // P_in2Heater
      const float dTin = (sol - Pw - Ph) * a1;
      const float dTh  = (heat + Ph) * a2;
      const float dTe  = (Pw + (Toutv - Te) * iR3) * a3;

      Tin += tame(dTin);
      Th  += tame(dTh);
      Te  += tame(dTe);

      if (active) {
        // Streaming stores: 12B/thread, contiguous across the wave (384B),
        // non-temporal so the 430MB trajectory doesn't thrash L2.
        __builtin_nontemporal_store(Tin, outp + 0);
        __builtin_nontemporal_store(Th,  outp + 1);
        __builtin_nontemporal_store(Te,  outp + 2);
      }
      outp += ostride;
    }
    __syncthreads();   // protect buffer (buf) before it is DMA-overwritten
  }
}

extern "C" void kernel_launch(void* const* d_in, const int* in_sizes, int n_in,
                              void* d_out, int out_size, void* d_ws, size_t ws_size,
                              hipStream_t stream) {
  const float* state0 = (const float*)d_in[0];   // (B,3)  fp32
  const float* params = (const float*)d_in[1];   // (B,6)  fp32
  const float* ext    = (const float*)d_in[2];   // (T,3)  fp32
  const int*   dtp    = (const int*)d_in[3];     // scalar int

  const int B = in_sizes[0] / 3;
  const int T = in_sizes[2] / 3;

  const int blocks = (B + NTHREADS - 1) / NTHREADS;
  rc3_scan_kernel<<<blocks, NTHREADS, 0, stream>>>(
      state0, params, ext, dtp, (float*)d_out, B, T);
}